// SegmentFeatureFusion_18373870092759
// MI455X (gfx1250) — compile-verified
//
#include <hip/hip_runtime.h>
#include <cstddef>

typedef __attribute__((ext_vector_type(2))) float v2f;
typedef __attribute__((ext_vector_type(4))) float v4f;
typedef __attribute__((ext_vector_type(8))) float v8f;

#define BATCH 2
#define CCH   64
#define SEGN  9
#define SPAD  16
#define HIDN  32
#define MORPHN 45
#define VOX   (64 * 128 * 128)       // 1,048,576 voxels per (b, channel/segment) plane
#define NBLK  512                    // blocks per batch
#define NWAVE 8                      // waves per block (256 threads, wave32)
#define VPW   (VOX / (NBLK * NWAVE)) // 256 voxels per wave
#define ITERS (VPW / 8)              // 8 voxels per iteration

__device__ __forceinline__ float fast_sigmoid(float x, float keep) {
    // keep * v_rcp_f32(1 + exp(-x)) : TRANS ops co-execute with WMMA,
    // avoids the v_div_scale/v_div_fmas/v_div_fixup IEEE-divide sequence.
    return keep * __builtin_amdgcn_rcpf(1.0f + __expf(-x));
}

// ---------------------------------------------------------------------------
// Kernel 0: zero the workspace accumulators (sums[B][16][64] + denom[B][16]).
// ---------------------------------------------------------------------------
__global__ void SegFuse_zero_ws(float* ws, int n) {
    int i = blockIdx.x * blockDim.x + threadIdx.x;
    if (i < n) ws[i] = 0.0f;
}

// ---------------------------------------------------------------------------
// Kernel 1: fused sigmoid + masked pooling via V_WMMA_F32_16X16X4_F32.
//
// GEMM view per batch:  sums[s, c] = sum_v  sigmoid(seg[s, v]) * feat[c, v]
//   A (16x4):  rows = segments (9 real + 7 zero-padded), K = voxels
//   B (4x16):  K = voxels, cols = 16 channels (4 tiles cover C=64)
//   C (16x16): f32 accumulators, 4 tiles x 8 VGPRs per lane
//
// A layout (32-bit 16x4): lane L<16 holds K=0,1 ; lane L>=16 holds K=2,3,
// row M = L%16.  B layout mirrors this for its K slots.  Because WMMA sums
// over K, each lane loads one aligned B128 (4 consecutive voxels at
// v + 4*(L/16)) and feeds .xy to the first WMMA and .zw to the second —
// the K-slot/voxel permutation is identical for A and B, so the sum is exact.
//
// Each wave walks a 1 KB window per row, so after the per-lane base pointers
// are formed, every load is base + constant immediate offset (fits IOFFSET).
// ---------------------------------------------------------------------------
__global__ __launch_bounds__(256) void SegFuse_pool_wmma(
        const float* __restrict__ feat,   // [B][C][V]
        const float* __restrict__ seg,    // [B][S][V]
        float* __restrict__ ws_sums,      // [B][16][64]
        float* __restrict__ ws_den)       // [B][16]
{
    __shared__ float lsum[NWAVE][SPAD * CCH];   // 8 * 1024 floats = 32 KB
    __shared__ float lden[NWAVE][32];

    const int tid  = threadIdx.x;
    const int lane = tid & 31;
    const int wv   = tid >> 5;
    const int b    = blockIdx.y;
    const int sr   = lane & 15;   // A row (segment) / B column (channel % 16)
    const int hi   = lane >> 4;   // 0: K-slots 0,1   1: K-slots 2,3

    const size_t v0 = ((size_t)blockIdx.x * NWAVE + wv) * (size_t)VPW + (size_t)hi * 4;

    // Clamp padded segment rows into bounds; their contribution is zeroed.
    const float* prow = seg + ((size_t)b * SEGN + (sr < SEGN ? sr : SEGN - 1)) * (size_t)VOX + v0;
    const float* f0 = feat + ((size_t)b * CCH +  0 + sr) * (size_t)VOX + v0;
    const float* f1 = feat + ((size_t)b * CCH + 16 + sr) * (size_t)VOX + v0;
    const float* f2 = feat + ((size_t)b * CCH + 32 + sr) * (size_t)VOX + v0;
    const float* f3 = feat + ((size_t)b * CCH + 48 + sr) * (size_t)VOX + v0;

    v8f acc0 = {}, acc1 = {}, acc2 = {}, acc3 = {};
    float dsum = 0.0f;
    const float keep = (sr < SEGN) ? 1.0f : 0.0f;   // select, not branch: EXEC stays full

#pragma unroll 4
    for (int i = 0; i < ITERS; ++i) {
        const int v = i * 8;   // constant byte offsets after unroll (max 248*4 B)

        // A operand: 4 sigmoids from one B128 load
        v4f pl = *(const v4f*)(prow + v);
        float s0 = fast_sigmoid(pl.x, keep);
        float s1 = fast_sigmoid(pl.y, keep);
        float s2 = fast_sigmoid(pl.z, keep);
        float s3 = fast_sigmoid(pl.w, keep);
        dsum += (s0 + s1) + (s2 + s3);
        v2f A0 = {s0, s1};
        v2f A1 = {s2, s3};

        // B operands: one B128 per channel tile, two WMMAs each
        v4f q0 = *(const v4f*)(f0 + v);
        v4f q1 = *(const v4f*)(f1 + v);
        v4f q2 = *(const v4f*)(f2 + v);
        v4f q3 = *(const v4f*)(f3 + v);

        v2f B0a = {q0.x, q0.y}, B0b = {q0.z, q0.w};
        v2f B1a = {q1.x, q1.y}, B1b = {q1.z, q1.w};
        v2f B2a = {q2.x, q2.y}, B2b = {q2.z, q2.w};
        v2f B3a = {q3.x, q3.y}, B3b = {q3.z, q3.w};

        acc0 = __builtin_amdgcn_wmma_f32_16x16x4_f32(false, A0, false, B0a, (short)0, acc0, false, false);
        acc0 = __builtin_amdgcn_wmma_f32_16x16x4_f32(false, A1, false, B0b, (short)0, acc0, false, false);
        acc1 = __builtin_amdgcn_wmma_f32_16x16x4_f32(false, A0, false, B1a, (short)0, acc1, false, false);
        acc1 = __builtin_amdgcn_wmma_f32_16x16x4_f32(false, A1, false, B1b, (short)0, acc1, false, false);
        acc2 = __builtin_amdgcn_wmma_f32_16x16x4_f32(false, A0, false, B2a, (short)0, acc2, false, false);
        acc2 = __builtin_amdgcn_wmma_f32_16x16x4_f32(false, A1, false, B2b, (short)0, acc2, false, false);
        acc3 = __builtin_amdgcn_wmma_f32_16x16x4_f32(false, A0, false, B3a, (short)0, acc3, false, false);
        acc3 = __builtin_amdgcn_wmma_f32_16x16x4_f32(false, A1, false, B3b, (short)0, acc3, false, false);
    }

    // D layout (16x16 f32): VGPR j, lane L  ->  M = j + 8*(L/16), N = L%16
#pragma unroll
    for (int j = 0; j < 8; ++j) {
        const int m = j + 8 * hi;
        lsum[wv][m * CCH +  0 + sr] = acc0[j];
        lsum[wv][m * CCH + 16 + sr] = acc1[j];
        lsum[wv][m * CCH + 32 + sr] = acc2[j];
        lsum[wv][m * CCH + 48 + sr] = acc3[j];
    }
    lden[wv][lane] = dsum;
    __syncthreads();

    // Cross-wave reduce then one atomic set per block into the global accumulator.
    for (int k = tid; k < SPAD * CCH; k += 256) {
        float r = 0.0f;
#pragma unroll
        for (int w = 0; w < NWAVE; ++w) r += lsum[w][k];
        atomicAdd(&ws_sums[b * SPAD * CCH + k], r);
    }
    if (tid < 32) {
        float r = 0.0f;
#pragma unroll
        for (int w = 0; w < NWAVE; ++w) r += lden[w][tid];
        atomicAdd(&ws_den[b * SPAD + (tid & 15)], r);  // lane and lane+16 fold to same segment
    }
}

// ---------------------------------------------------------------------------
// Kernel 2: tiny epilogue — pooled = sums/clip(denom), seg_proj + morph_proj
// + 2-layer classifier.  ~60K MACs total; 1 block of 64 threads (b = t/32,
// h = t%32).
// ---------------------------------------------------------------------------
__global__ void SegFuse_epilogue(
        const float* __restrict__ ws_sums, const float* __restrict__ ws_den,
        const float* __restrict__ morph,
        const float* __restrict__ w_seg,  const float* __restrict__ b_seg,
        const float* __restrict__ w_morph,const float* __restrict__ b_morph,
        const float* __restrict__ w_cls1, const float* __restrict__ b_cls1,
        const float* __restrict__ w_cls2, const float* __restrict__ b_cls2,
        float* __restrict__ out)
{
    __shared__ float fused[BATCH][(SEGN + 1) * HIDN];  // 2 x 320
    __shared__ float hdn[BATCH][HIDN];

    const int t = threadIdx.x;           // 0..63
    const int b = t >> 5;
    const int h = t & 31;

    for (int si = 0; si < SEGN; ++si) {
        float dn = ws_den[b * SPAD + si];
        dn = dn < 1.0f ? 1.0f : dn;
        float inv = 1.0f / dn;           // cold path: precise divide is fine
        float acc = b_seg[h];
        for (int c = 0; c < CCH; ++c)
            acc += (ws_sums[(b * SPAD + si) * CCH + c] * inv) * w_seg[c * HIDN + h];
        fused[b][si * HIDN + h] = acc > 0.0f ? acc : 0.0f;
    }
    {
        float acc = b_morph[h];
        for (int m = 0; m < MORPHN; ++m)
            acc += morph[b * MORPHN + m] * w_morph[m * HIDN + h];
        fused[b][SEGN * HIDN + h] = acc > 0.0f ? acc : 0.0f;
    }
    __syncthreads();
    {
        float acc = b_cls1[h];
        for (int k = 0; k < (SEGN + 1) * HIDN; ++k)
            acc += fused[b][k] * w_cls1[k * HIDN + h];
        hdn[b][h] = acc > 0.0f ? acc : 0.0f;
    }
    __syncthreads();
    if (h == 0) {
        float acc = b_cls2[0];
        for (int k = 0; k < HIDN; ++k) acc += hdn[b][k] * w_cls2[k];
        out[b] = acc;
    }
}

extern "C" void kernel_launch(void* const* d_in, const int* in_sizes, int n_in,
                              void* d_out, int out_size, void* d_ws, size_t ws_size,
                              hipStream_t stream) {
    const float* feat    = (const float*)d_in[0];
    const float* seg     = (const float*)d_in[1];
    const float* morph   = (const float*)d_in[2];
    const float* w_seg   = (const float*)d_in[3];
    const float* b_seg   = (const float*)d_in[4];
    const float* w_morph = (const float*)d_in[5];
    const float* b_morph = (const float*)d_in[6];
    const float* w_cls1  = (const float*)d_in[7];
    const float* b_cls1  = (const float*)d_in[8];
    const float* w_cls2  = (const float*)d_in[9];
    const float* b_cls2  = (const float*)d_in[10];
    float* out = (float*)d_out;

    float* ws_sums = (float*)d_ws;                        // B*16*64 = 2048 floats
    float* ws_den  = ws_sums + BATCH * SPAD * CCH;        // B*16    = 32 floats
    const int nacc = BATCH * SPAD * CCH + BATCH * SPAD;   // 2080

    SegFuse_zero_ws<<<(nacc + 255) / 256, 256, 0, stream>>>((float*)d_ws, nacc);

    dim3 grid(NBLK, BATCH);
    SegFuse_pool_wmma<<<grid, 256, 0, stream>>>(feat, seg, ws_sums, ws_den);

    SegFuse_epilogue<<<1, 64, 0, stream>>>(ws_sums, ws_den, morph,
                                           w_seg, b_seg, w_morph, b_morph,
                                           w_cls1, b_cls1, w_cls2, b_cls2, out);
}